// Tree_Net_15960098472813
// MI455X (gfx1250) — compile-verified
//
#include <hip/hip_runtime.h>
#include <hip/hip_bf16.h>
#include <math.h>

// ---------------------------------------------------------------------------
// Problem constants (from the reference)
// ---------------------------------------------------------------------------
static constexpr int N_LEAVES   = 8192;
static constexpr int N_INTERNAL = N_LEAVES - 1;          // 8191
static constexpr int N_NODES    = N_LEAVES + N_INTERNAL; // 16383
static constexpr int DIM        = 512;
static constexpr int N_CAT      = 512;

// ---------------------------------------------------------------------------
// CDNA5 fp32 WMMA: V_WMMA_F32_16X16X4_F32  (D = A(16x4) * B(4x16) + C(16x16))
// NOTE: __has_builtin only sees the *current* target's builtins; amdgcn
// builtins are aux-target builtins during the host pass, so only sanity-check
// on the device pass.
// ---------------------------------------------------------------------------
typedef __attribute__((ext_vector_type(2))) float v2f;
typedef __attribute__((ext_vector_type(8))) float v8f;

#if defined(__HIP_DEVICE_COMPILE__) && !__has_builtin(__builtin_amdgcn_wmma_f32_16x16x4_f32)
#error "gfx1250 device pass is missing __builtin_amdgcn_wmma_f32_16x16x4_f32"
#endif

__device__ __forceinline__ float waveReduceSum(float v) {
  // wave32 butterfly reduction
  #pragma unroll
  for (int off = 16; off > 0; off >>= 1) v += __shfl_xor(v, off, 32);
  return v;
}

// ---------------------------------------------------------------------------
// Kernel 1: gather leaf embeddings, L2-normalize. One wave (32 lanes) per leaf,
// 16 floats per lane as 4x float4.
// ---------------------------------------------------------------------------
__global__ void leaf_normalize_kernel(const int* __restrict__ ids,
                                      const float* __restrict__ emb,
                                      float* __restrict__ nodes) {
  const int wavesPerBlock = blockDim.x >> 5;
  const int leaf = blockIdx.x * wavesPerBlock + (threadIdx.x >> 5);
  const int lane = threadIdx.x & 31;
  if (leaf >= N_LEAVES) return;

  const float4* src = reinterpret_cast<const float4*>(emb + (size_t)ids[leaf] * DIM);
  float4 v[4];
  float ss = 0.0f;
  #pragma unroll
  for (int i = 0; i < 4; ++i) {
    v[i] = src[lane + 32 * i];
    ss += v[i].x * v[i].x + v[i].y * v[i].y + v[i].z * v[i].z + v[i].w * v[i].w;
  }
  ss = waveReduceSum(ss);
  const float rn = 1.0f / sqrtf(ss);

  float4* dst = reinterpret_cast<float4*>(nodes + (size_t)leaf * DIM);
  #pragma unroll
  for (int i = 0; i < 4; ++i) {
    float4 o;
    o.x = v[i].x * rn; o.y = v[i].y * rn; o.z = v[i].z * rn; o.w = v[i].w * rn;
    dst[lane + 32 * i] = o;
  }
}

// ---------------------------------------------------------------------------
// Kernel 2: one tree level. One 256-thread block per internal node.
//   v[n] = sum_j a[j] * c[(j+n) mod 512], then v /= ||v||.
// a broadcast from LDS (conflict-free), c duplicated (1024 floats) to kill
// the modulo; lane-consecutive n => conflict-free LDS reads.
// ---------------------------------------------------------------------------
__global__ void tree_level_kernel(float* __restrict__ nodes,
                                  const int* __restrict__ left,
                                  const int* __restrict__ right,
                                  int kStart) {
  __shared__ float a_s[DIM];
  __shared__ float c_s[2 * DIM];
  __shared__ float red_s[8];
  __shared__ float rnorm_s;

  const int k = kStart + blockIdx.x;          // internal node index
  const int t = threadIdx.x;                  // 256 threads
  const int l = left[k];
  const int r = right[k];

  const float* aP = nodes + (size_t)l * DIM;
  const float* cP = nodes + (size_t)r * DIM;
  for (int i = t; i < DIM; i += 256) {
    const float cv = cP[i];
    a_s[i] = aP[i];
    c_s[i] = cv;
    c_s[i + DIM] = cv;
  }
  __syncthreads();

  const int n0 = t, n1 = t + 256;
  float acc0 = 0.0f, acc1 = 0.0f;
  #pragma unroll 8
  for (int j = 0; j < DIM; ++j) {
    const float aj = a_s[j];
    acc0 = fmaf(aj, c_s[j + n0], acc0);
    acc1 = fmaf(aj, c_s[j + n1], acc1);
  }

  // block-wide sum of squares -> 1/norm
  float ss = waveReduceSum(acc0 * acc0 + acc1 * acc1);
  const int wave = t >> 5, lane = t & 31;
  if (lane == 0) red_s[wave] = ss;
  __syncthreads();
  if (t == 0) {
    float tot = 0.0f;
    #pragma unroll
    for (int w = 0; w < 8; ++w) tot += red_s[w];
    rnorm_s = 1.0f / sqrtf(tot);
  }
  __syncthreads();

  const float rn = rnorm_s;
  float* dst = nodes + (size_t)(N_LEAVES + k) * DIM;
  dst[n0] = acc0 * rn;
  dst[n1] = acc1 * rn;
}

// ---------------------------------------------------------------------------
// Kernel 3: logits = nodes(16383x512) @ W^T(512x512) + b  via fp32 WMMA.
// One wave per 16x16 output tile, K-loop step 4 (V_WMMA_F32_16X16X4_F32).
//
// Fragment layouts (ISA 7.12.2, 32-bit):
//   A 16x4 : lane L -> row L%16; VGPR0/1 = K = 2*(L/16) + {0,1}  -> float2 load
//   B 4x16 : lane L -> col L%16; VGPR0/1 = K = 2*(L/16) + {0,1}  (symmetric)
//   C 16x16: VGPR v -> row v + 8*(L/16), col L%16
// B[k,n] = W[n,k], so both fragments are contiguous float2 global loads.
// ---------------------------------------------------------------------------
__global__ void logits_wmma_kernel(const float* __restrict__ nodes,
                                   const float* __restrict__ W,
                                   const float* __restrict__ bias,
                                   float* __restrict__ out) {
  const int lane   = threadIdx.x & 31;
  const int waveId = threadIdx.x >> 5;
  const int tile   = blockIdx.x * (blockDim.x >> 5) + waveId; // 0..32767
  const int mtile  = tile >> 5;  // 0..1023  (rows of 16)
  const int ntile  = tile & 31;  // 0..31    (cols of 16)

  const int m16  = lane & 15;
  const int half = lane >> 4;

  const int row  = mtile * 16 + m16;
  const int rowc = row < N_NODES ? row : (N_NODES - 1);   // clamp last tile
  const int col  = ntile * 16 + m16;

  const float* aBase = nodes + (size_t)rowc * DIM + 2 * half;
  const float* bBase = W     + (size_t)col  * DIM + 2 * half;

  v8f acc = {};
  #pragma unroll 4
  for (int k0 = 0; k0 < DIM; k0 += 4) {
    const v2f a = *reinterpret_cast<const v2f*>(aBase + k0);
    const v2f b = *reinterpret_cast<const v2f*>(bBase + k0);
    // 8 args: (neg_a, A, neg_b, B, c_mod, C, reuse_a, reuse_b)
    acc = __builtin_amdgcn_wmma_f32_16x16x4_f32(false, a, false, b,
                                                (short)0, acc, false, false);
  }

  const float bv = bias[col];
  #pragma unroll
  for (int v = 0; v < 8; ++v) {
    const int orow = mtile * 16 + v + 8 * half;
    if (orow < N_NODES) out[(size_t)orow * N_CAT + col] = acc[v] + bv;
  }
}

// ---------------------------------------------------------------------------
// Kernel 4: in-place row softmax over 512 categories. One 512-thread block/row.
// ---------------------------------------------------------------------------
__global__ void softmax_kernel(float* __restrict__ out) {
  __shared__ float red[N_CAT];
  const int row = blockIdx.x;
  const int t   = threadIdx.x;
  float* p = out + (size_t)row * N_CAT;

  const float x = p[t];
  red[t] = x;
  __syncthreads();
  for (int s = 256; s > 0; s >>= 1) {
    if (t < s) red[t] = fmaxf(red[t], red[t + s]);
    __syncthreads();
  }
  const float m = red[0];
  __syncthreads();

  const float e = expf(x - m);
  red[t] = e;
  __syncthreads();
  for (int s = 256; s > 0; s >>= 1) {
    if (t < s) red[t] += red[t + s];
    __syncthreads();
  }
  p[t] = e / red[0];
}

// ---------------------------------------------------------------------------
// Host-side launch. d_ws holds the node buffer: 16383*512 floats (~33.6 MB).
// Stream ordering provides the level-by-level dependency of the tree climb.
// ---------------------------------------------------------------------------
extern "C" void kernel_launch(void* const* d_in, const int* in_sizes, int n_in,
                              void* d_out, int out_size, void* d_ws, size_t ws_size,
                              hipStream_t stream) {
  const int*   content_ids = (const int*)  d_in[0];
  const int*   left_idx    = (const int*)  d_in[1];
  const int*   right_idx   = (const int*)  d_in[2];
  const float* emb         = (const float*)d_in[3];
  const float* W           = (const float*)d_in[4];
  const float* bias        = (const float*)d_in[5];
  float*       out         = (float*)d_out;
  float*       nodes       = (float*)d_ws;   // [N_NODES, DIM] scratch

  // 1) leaves: 8192 waves, 8 waves per 256-thread block
  leaf_normalize_kernel<<<N_LEAVES / 8, 256, 0, stream>>>(content_ids, emb, nodes);

  // 2) 13 tree levels (level l: count = 8192>>l internal nodes, heap layout)
  for (int lev = 1; lev <= 13; ++lev) {
    const int count  = N_LEAVES >> lev;
    const int kStart = N_LEAVES - ((2 * N_LEAVES) >> lev);
    tree_level_kernel<<<count, 256, 0, stream>>>(nodes, left_idx, right_idx, kStart);
  }

  // 3) fp32-WMMA projection: 1024 x 32 = 32768 tiles, 8 waves per block
  logits_wmma_kernel<<<32768 / 8, 256, 0, stream>>>(nodes, W, bias, out);

  // 4) softmax per node row, in place in d_out
  softmax_kernel<<<N_NODES, N_CAT, 0, stream>>>(out);
}